// GlobalRouter_78606491451537
// MI455X (gfx1250) — compile-verified
//
#include <hip/hip_runtime.h>
#include <math.h>

#define TOKENS   16384
#define HIDDEN   4096
#define NEXP     64
#define TOPK     2
#define AUX_COEF 0.01f

// d_out layout (floats, reference tuple flattened in order):
//   [0 .. 32767]            expert_indices (as float)
//   [32768 .. 65535]        scores
//   [65536 .. 2162687]      expert_mask [tokens][2][64]
//   [2162688]               aux_loss
#define OUT_IDX   0
#define OUT_SCORE (TOKENS * TOPK)
#define OUT_MASK  (2 * TOKENS * TOPK)
#define OUT_AUX   (OUT_MASK + (size_t)TOKENS * TOPK * NEXP)

typedef __attribute__((ext_vector_type(2))) float v2f;
typedef __attribute__((ext_vector_type(8))) float v8f;

__global__ void moe_init(float* __restrict__ ws) {
    int i = threadIdx.x;
    if (i < 128) ws[i] = 0.0f;  // gF[64] | gM[64]
}

__global__ __launch_bounds__(32)
void moe_router(const float* __restrict__ x,
                const float* __restrict__ Wg,
                const float* __restrict__ bg,
                float* __restrict__ out,
                float* __restrict__ gF,
                float* __restrict__ gM) {
    __shared__ float lg[16 * 65];   // logits tile, padded stride 65 -> conflict-free row scans
    __shared__ float mAcc[64];      // per-block softmax-prob accumulation
    __shared__ int   idxS[32];      // top-2 indices for 16 tokens

    const int lane = threadIdx.x;       // 0..31
    const int r    = lane & 15;
    const int h    = lane >> 4;         // 0: K pair {0,1}, 1: K pair {2,3}
    const int tokenBase = blockIdx.x * 16;

    v8f acc0 = {}, acc1 = {}, acc2 = {}, acc3 = {};

    // A: lane r (<16) = row M=r; VGPR0 = x[M][k0+2h], VGPR1 = x[M][k0+2h+1]
    const float* xrow  = x + (size_t)(tokenBase + r) * HIDDEN + 2 * h;
    // B: 4x16 tile; lane layout mirrors A: VGPR0 = W[k0+2h][n], VGPR1 = W[k0+2h+1][n]
    const float* wbase = Wg + r;

    for (int kc = 0; kc < HIDDEN; kc += 32) {
        __builtin_prefetch(xrow + kc + 32, 0, 0);   // global_prefetch_b8, next x chunk
#pragma unroll
        for (int k0 = kc; k0 < kc + 32; k0 += 4) {
            v2f a = *(const v2f*)(xrow + k0);       // 8B aligned contiguous pair
            const float* w0 = wbase + (size_t)(k0 + 2 * h) * NEXP;
            v2f b0, b1, b2, b3;
            b0.x = w0[0];   b0.y = w0[NEXP + 0];
            b1.x = w0[16];  b1.y = w0[NEXP + 16];
            b2.x = w0[32];  b2.y = w0[NEXP + 32];
            b3.x = w0[48];  b3.y = w0[NEXP + 48];
            acc0 = __builtin_amdgcn_wmma_f32_16x16x4_f32(false, a, false, b0, (short)0, acc0, false, false);
            acc1 = __builtin_amdgcn_wmma_f32_16x16x4_f32(false, a, false, b1, (short)0, acc1, false, false);
            acc2 = __builtin_amdgcn_wmma_f32_16x16x4_f32(false, a, false, b2, (short)0, acc2, false, false);
            acc3 = __builtin_amdgcn_wmma_f32_16x16x4_f32(false, a, false, b3, (short)0, acc3, false, false);
        }
    }

    // Bias depends on N only; lane's column within each tile is r.
    const float bb0 = bg[r], bb1 = bg[r + 16], bb2 = bg[r + 32], bb3 = bg[r + 48];

    // C/D layout: lanes 0-15 hold M=0..7 (VGPR j -> M=j), lanes 16-31 hold M=8..15.
#pragma unroll
    for (int j = 0; j < 8; ++j) {
        const int m = h * 8 + j;
        lg[m * 65 + r +  0] = acc0[j] + bb0;
        lg[m * 65 + r + 16] = acc1[j] + bb1;
        lg[m * 65 + r + 32] = acc2[j] + bb2;
        lg[m * 65 + r + 48] = acc3[j] + bb3;
    }
    mAcc[lane]      = 0.0f;
    mAcc[lane + 32] = 0.0f;
    __syncthreads();

    if (lane < 16) {
        const float* row = lg + lane * 65;
        float v1 = row[0]; int i1 = 0;
        float v2 = -1e30f; int i2 = 0;
        float mx = v1;
#pragma unroll 4
        for (int e = 1; e < NEXP; ++e) {
            float v = row[e];
            mx = fmaxf(mx, v);
            if (v > v1)      { v2 = v1; i2 = i1; v1 = v; i1 = e; }
            else if (v > v2) { v2 = v;  i2 = e; }
        }
        float denom = 0.0f;
#pragma unroll 4
        for (int e = 0; e < NEXP; ++e) denom += __expf(row[e] - mx);
        const float inv = 1.0f / denom;
#pragma unroll 4
        for (int e = 0; e < NEXP; ++e)
            atomicAdd(&mAcc[e], __expf(row[e] - mx) * inv);   // ds_add_f32

        // softmax over the two selected logits (v1 >= v2)
        const float s0 = 1.0f / (1.0f + __expf(v2 - v1));
        const int tok = tokenBase + lane;
        out[OUT_IDX   + tok * 2 + 0] = (float)i1;
        out[OUT_IDX   + tok * 2 + 1] = (float)i2;
        out[OUT_SCORE + tok * 2 + 0] = s0;
        out[OUT_SCORE + tok * 2 + 1] = 1.0f - s0;
        idxS[lane * 2 + 0] = i1;
        idxS[lane * 2 + 1] = i2;
        atomicAdd(&gF[i1], 1.0f);
        atomicAdd(&gF[i2], 1.0f);
    }
    __syncthreads();

    // One-hot expert_mask: wave-cooperative fill, 128 floats per token.
    for (int tt = 0; tt < 16; ++tt) {
        const int tok = tokenBase + tt;
        const int j1 = idxS[tt * 2 + 0];
        const int j2 = idxS[tt * 2 + 1];
        float* mrow = out + OUT_MASK + (size_t)tok * (TOPK * NEXP);
#pragma unroll
        for (int j = lane; j < TOPK * NEXP; j += 32) {
            const int k = j >> 6;
            const int e = j & 63;
            mrow[j] = (e == (k ? j2 : j1)) ? 1.0f : 0.0f;
        }
    }

    // Per-block m_i partials -> global (2 atomics per lane)
    atomicAdd(&gM[lane],      mAcc[lane]);
    atomicAdd(&gM[lane + 32], mAcc[lane + 32]);
}

__global__ void moe_final(const float* __restrict__ gF,
                          const float* __restrict__ gM,
                          float* __restrict__ out) {
    if (threadIdx.x == 0) {
        float s = 0.0f;
        for (int e = 0; e < NEXP; ++e) {
            const float fi = gF[e] * (1.0f / ((float)TOKENS * (float)TOPK));
            const float mi = gM[e] * (1.0f / (float)TOKENS);
            s += fi * mi;
        }
        out[OUT_AUX] = AUX_COEF * s / (float)NEXP;
    }
}

extern "C" void kernel_launch(void* const* d_in, const int* in_sizes, int n_in,
                              void* d_out, int out_size, void* d_ws, size_t ws_size,
                              hipStream_t stream) {
    const float* x  = (const float*)d_in[0];
    const float* Wg = (const float*)d_in[1];
    const float* bg = (const float*)d_in[2];
    float* out = (float*)d_out;
    float* ws  = (float*)d_ws;          // gF = ws[0..63], gM = ws[64..127]

    moe_init<<<1, 128, 0, stream>>>(ws);
    moe_router<<<TOKENS / 16, 32, 0, stream>>>(x, Wg, bg, out, ws, ws + 64);
    moe_final<<<1, 32, 0, stream>>>(ws, ws + 64, out);
}